// VanillaRateRNN_17746804867992
// MI455X (gfx1250) — compile-verified
//
#include <hip/hip_runtime.h>
#include <cmath>

typedef float v2f __attribute__((ext_vector_type(2)));
typedef float v8f __attribute__((ext_vector_type(8)));

#define RNN_ALPHA 0.5f

constexpr int Bsz = 64, T = 1024, I = 64, H = 512, O = 32;
constexpr int HSTR = H + 4;   // 516: conflict-free LDS stride (516 % 64 == 4)
constexpr int XSTR = I + 4;   // 68

__device__ __forceinline__ v8f wmma4(v2f a, v2f b, v8f c) {
  // fp32 WMMA: D = A(16x4) * B(4x16) + C(16x16), full fp32 precision
  return __builtin_amdgcn_wmma_f32_16x16x4_f32(false, a, false, b, (short)0, c,
                                               false, false);
}

__device__ __forceinline__ float fast_tanh(float v) {
#if __has_builtin(__builtin_amdgcn_tanhf)
  return __builtin_amdgcn_tanhf(v);
#else
  return tanhf(v);
#endif
}

// ---------------------------------------------------------------------------
// Phase 1+2 fused: sequential scan. Grid = 4 blocks (16 batch rows each),
// 512 threads = 16 waves; wave w owns output columns [32w, 32w+32).
// h state lives in LDS; W_hh / W_xh streamed from L2 every step.
// ---------------------------------------------------------------------------
__global__ __launch_bounds__(512) void rnn_scan_kernel(
    const float* __restrict__ x, const float* __restrict__ h0,
    const float* __restrict__ W_xh, const float* __restrict__ W_hh,
    const float* __restrict__ bias, float* __restrict__ h_all) {
  __shared__ float hbuf[16 * HSTR];
  __shared__ float xbuf[16 * XSTR];

  const int tid  = threadIdx.x;
  const int lane = tid & 31;
  const int wv   = tid >> 5;         // wave 0..15
  const int l15  = lane & 15;
  const int kh   = lane >> 4;        // lane-half selects K pair {0,1} vs {2,3}
  const int b0   = blockIdx.x * 16;  // batch row base
  const int n0   = wv * 32;          // wave's first output column

  // load h0 tile into LDS
  for (int i = tid; i < 16 * H; i += 512) {
    int r = i >> 9, c = i & (H - 1);
    hbuf[r * HSTR + c] = h0[(size_t)(b0 + r) * H + c];
  }

  const float bias0 = bias[n0 + l15];
  const float bias1 = bias[n0 + 16 + l15];

  // per-lane B-fragment row pointers: B[k][n] = W[n][k], row n contiguous in k
  const float* Bhh0 = W_hh + (size_t)(n0 + l15) * H + kh * 2;
  const float* Bhh1 = W_hh + (size_t)(n0 + 16 + l15) * H + kh * 2;
  const float* Bxh0 = W_xh + (size_t)(n0 + l15) * I + kh * 2;
  const float* Bxh1 = W_xh + (size_t)(n0 + 16 + l15) * I + kh * 2;

  const int xr = tid >> 5;           // x-staging row
  const int xi = (tid & 31) * 2;     // x-staging column pair

  for (int t = 0; t < T; ++t) {
    // stage x[:, t, :] (16 x 64) into LDS
    {
      const float* src = x + ((size_t)(b0 + xr) * T + t) * I + xi;
      *(v2f*)&xbuf[xr * XSTR + xi] = *(const v2f*)src;
    }
    __syncthreads();  // hbuf(t-1) stores + xbuf(t) visible

    v8f acc0, acc1;
#pragma unroll
    for (int v = 0; v < 8; ++v) { acc0[v] = bias0; acc1[v] = bias1; }

    // recurrent GEMM contribution: pre += h @ W_hh^T   (K = 512)
#pragma unroll 8
    for (int kk = 0; kk < H; kk += 4) {
      v2f a  = *(const v2f*)&hbuf[l15 * HSTR + kk + kh * 2];  // ds_load_b64
      v2f w0 = *(const v2f*)(Bhh0 + kk);                      // global b64 (L2)
      v2f w1 = *(const v2f*)(Bhh1 + kk);
      acc0 = wmma4(a, w0, acc0);
      acc1 = wmma4(a, w1, acc1);
    }
    // input projection contribution: pre += x_t @ W_xh^T  (K = 64)
#pragma unroll
    for (int kk = 0; kk < I; kk += 4) {
      v2f a  = *(const v2f*)&xbuf[l15 * XSTR + kk + kh * 2];
      v2f w0 = *(const v2f*)(Bxh0 + kk);
      v2f w1 = *(const v2f*)(Bxh1 + kk);
      acc0 = wmma4(a, w0, acc0);
      acc1 = wmma4(a, w1, acc1);
    }

    // h_new = (1-a)*h + a*tanh(pre); C/D layout: m = v + 8*kh, n = n0 + l15
    float hn0[8], hn1[8];
#pragma unroll
    for (int v = 0; v < 8; ++v) {
      int m = v + 8 * kh;
      float o0 = hbuf[m * HSTR + n0 + l15];
      float o1 = hbuf[m * HSTR + n0 + 16 + l15];
      hn0[v] = (1.0f - RNN_ALPHA) * o0 + RNN_ALPHA * fast_tanh(acc0[v]);
      hn1[v] = (1.0f - RNN_ALPHA) * o1 + RNN_ALPHA * fast_tanh(acc1[v]);
    }
    // write h_all (global) — no LDS hazard, safe before the barrier
#pragma unroll
    for (int v = 0; v < 8; ++v) {
      int m = v + 8 * kh;
      size_t rowbase = ((size_t)(b0 + m) * T + t) * (size_t)H;
      h_all[rowbase + n0 + l15]      = hn0[v];
      h_all[rowbase + n0 + 16 + l15] = hn1[v];
    }
    __syncthreads();  // everyone done reading h_old
#pragma unroll
    for (int v = 0; v < 8; ++v) {
      int m = v + 8 * kh;
      hbuf[m * HSTR + n0 + l15]      = hn0[v];
      hbuf[m * HSTR + n0 + 16 + l15] = hn1[v];
    }
  }
}

// ---------------------------------------------------------------------------
// Phase 3: y = h_all @ W_hy^T + b_hy. 8192 16x16 tiles, one per wave.
// ---------------------------------------------------------------------------
__global__ __launch_bounds__(256) void out_proj_kernel(
    const float* __restrict__ h_all, const float* __restrict__ W_hy,
    const float* __restrict__ b_hy, float* __restrict__ y) {
  const int lane = threadIdx.x & 31;
  const int wv   = threadIdx.x >> 5;
  const int tile = blockIdx.x * 8 + wv;  // 0..8191
  const int n0   = (tile & 1) * 16;
  const int m0   = (tile >> 1) * 16;     // flat (b,t) row tile
  const int l15  = lane & 15;
  const int kh   = lane >> 4;

  const float bv = b_hy[n0 + l15];
  v8f acc;
#pragma unroll
  for (int v = 0; v < 8; ++v) acc[v] = bv;

  const float* Arow = h_all + (size_t)(m0 + l15) * H + kh * 2;
  const float* Brow = W_hy + (size_t)(n0 + l15) * H + kh * 2;
#pragma unroll 8
  for (int kk = 0; kk < H; kk += 4) {
    v2f a = *(const v2f*)(Arow + kk);
    v2f b = *(const v2f*)(Brow + kk);
    acc = wmma4(a, b, acc);
  }
#pragma unroll
  for (int v = 0; v < 8; ++v) {
    y[(size_t)(m0 + v + 8 * kh) * O + n0 + l15] = acc[v];
  }
}

// ---------------------------------------------------------------------------
extern "C" void kernel_launch(void* const* d_in, const int* in_sizes, int n_in,
                              void* d_out, int out_size, void* d_ws,
                              size_t ws_size, hipStream_t stream) {
  (void)in_sizes; (void)n_in; (void)out_size; (void)d_ws; (void)ws_size;
  const float* x    = (const float*)d_in[0];  // (B,T,I)
  const float* h0   = (const float*)d_in[1];  // (B,H)
  const float* W_xh = (const float*)d_in[2];  // (H,I)
  const float* W_hh = (const float*)d_in[3];  // (H,H)
  const float* bias = (const float*)d_in[4];  // (H)
  const float* W_hy = (const float*)d_in[5];  // (O,H)
  const float* b_hy = (const float*)d_in[6];  // (O)

  float* y     = (float*)d_out;                        // (B,T,O)
  float* h_all = y + (size_t)Bsz * T * O;              // (B,T,H)

  rnn_scan_kernel<<<Bsz / 16, 512, 0, stream>>>(x, h0, W_xh, W_hh, bias, h_all);

  const int tiles = (Bsz * T / 16) * (O / 16);         // 8192
  out_proj_kernel<<<tiles / 8, 256, 0, stream>>>(h_all, W_hy, b_hy, y);
}